// FSRGraphConv_7687991460131
// MI455X (gfx1250) — compile-verified
//
#include <hip/hip_runtime.h>

#define N_NODES 50000
#define N_EDGES 800000
#define IN_F    128
#define E_F     32
#define OUT_F   128
#define CAT_F   (IN_F + E_F)      // 160
#define CAT2_F  (IN_F + OUT_F)    // 256 (W_w second dim)

typedef __attribute__((ext_vector_type(2))) float v2f;
typedef __attribute__((ext_vector_type(8))) float v8f;

// ---------------- Kernel 1: zero accumulators ----------------
__global__ void zero_ws_kernel(float* __restrict__ ws, int n) {
    int i = blockIdx.x * blockDim.x + threadIdx.x;
    if (i < n) ws[i] = 0.0f;
}

// ---------------- Kernel 2: edge scatter (segment sums) ----------------
// One wave32 per edge. Lanes stride the 128 h-features (4 passes) and the
// 32 edge-features (1 pass). Lane 0 bumps the degree count.
__global__ __launch_bounds__(256)
void edge_scatter_kernel(const float* __restrict__ h,
                         const float* __restrict__ eftr,
                         const int*   __restrict__ src_idx,
                         const int*   __restrict__ dst_idx,
                         float* __restrict__ sum_h,
                         float* __restrict__ sum_e,
                         float* __restrict__ cnt) {
    int wave = (blockIdx.x * blockDim.x + threadIdx.x) >> 5;
    int lane = threadIdx.x & 31;
    if (wave >= N_EDGES) return;

    int s = src_idx[wave];
    int d = dst_idx[wave];

    const float* hs  = h     + (size_t)s * IN_F;
    float*       shd = sum_h + (size_t)d * IN_F;
#pragma unroll
    for (int i = 0; i < IN_F; i += 32)
        atomicAdd(&shd[i + lane], hs[i + lane]);

    atomicAdd(&sum_e[(size_t)d * E_F + lane], eftr[(size_t)wave * E_F + lane]);

    if (lane == 0) atomicAdd(&cnt[d], 1.0f);
}

// ---------------- Kernel 3: fused dual GEMM via fp32 WMMA ----------------
// Block = 256 threads = 8 waves; block handles 16 nodes x 128 output cols.
// Wave w owns output column tile [16w, 16w+16).
//   GEMM1: h_neigh(16x128) = feat_mean(16x160) @ weight(160x128)
//   GEMM2: out(16x128)     = h_dst(16x128) @ W_w[:, :128]^T
//                          + h_neigh(16x128) @ W_w[:, 128:]^T + W_b + bias
__global__ __launch_bounds__(256)
void fused_gemm_kernel(const float* __restrict__ h,
                       const float* __restrict__ weight,  // (160,128) row-major
                       const float* __restrict__ W_w,     // (128,256) row-major
                       const float* __restrict__ W_b,
                       const float* __restrict__ bias,
                       const float* __restrict__ sum_h,
                       const float* __restrict__ sum_e,
                       const float* __restrict__ cnt,
                       float* __restrict__ out) {
    __shared__ float feat[16 * CAT_F];   // feat_mean tile (16x160)
    __shared__ float hdst[16 * IN_F];    // h_dst tile     (16x128)
    __shared__ float hnb [16 * OUT_F];   // h_neigh tile   (16x128)

    const int tid  = threadIdx.x;
    const int row0 = blockIdx.x * 16;    // N_NODES == 3125 * 16, exact

    // Stage feat_mean = [sum_h/denom | sum_e/denom] (2560 elems, 10/thread)
    for (int i = tid; i < 16 * CAT_F; i += 256) {
        int m = i / CAT_F;
        int k = i - m * CAT_F;
        int node = row0 + m;
        float denom = fmaxf(cnt[node], 1.0f);
        float v = (k < IN_F) ? sum_h[(size_t)node * IN_F + k]
                             : sum_e[(size_t)node * E_F + (k - IN_F)];
        feat[i] = v / denom;
    }
    // Stage h_dst tile (2048 elems, 8/thread)
    for (int i = tid; i < 16 * IN_F; i += 256) {
        int m = i >> 7;
        int k = i & 127;
        hdst[i] = h[(size_t)(row0 + m) * IN_F + k];
    }
    __syncthreads();

    const int wv   = tid >> 5;
    const int lane = tid & 31;
    // fp32 16x16x4 fragment indexing (ISA 7.12.2, 32-bit A 16x4):
    //   lanes 0-15: M=lane,    comp0->K=0, comp1->K=1
    //   lanes16-31: M=lane-16, comp0->K=2, comp1->K=3
    const int mA = lane & 15;
    const int kA = (lane >> 4) << 1;
    const int n  = wv * 16 + (lane & 15);   // output column 0..127

    // ---- GEMM1: feat(16x160) @ weight(160x128) -> hnb tile ----
    v8f acc = {};
    for (int k0 = 0; k0 < CAT_F; k0 += 4) {
        v2f a, b;
        a.x = feat[mA * CAT_F + k0 + kA];
        a.y = feat[mA * CAT_F + k0 + kA + 1];
        b.x = weight[(size_t)(k0 + kA)     * OUT_F + n];
        b.y = weight[(size_t)(k0 + kA + 1) * OUT_F + n];
        acc = __builtin_amdgcn_wmma_f32_16x16x4_f32(
            false, a, false, b, (short)0, acc, false, false);
    }
#pragma unroll
    for (int r = 0; r < 8; ++r) {
        int m = r + ((lane >> 4) << 3);      // C/D layout: rows r / r+8
        hnb[m * OUT_F + n] = acc[r];
    }
    __syncthreads();

    // ---- GEMM2: [hdst | hnb](16x256) @ W_w^T(256x128) ----
    v8f acc2 = {};
    for (int k0 = 0; k0 < IN_F; k0 += 4) {   // first 128 K: h_dst part
        v2f a, b;
        a.x = hdst[mA * IN_F + k0 + kA];
        a.y = hdst[mA * IN_F + k0 + kA + 1];
        b.x = W_w[(size_t)n * CAT2_F + k0 + kA];       // W_w^T[k][n] = W_w[n][k]
        b.y = W_w[(size_t)n * CAT2_F + k0 + kA + 1];
        acc2 = __builtin_amdgcn_wmma_f32_16x16x4_f32(
            false, a, false, b, (short)0, acc2, false, false);
    }
    for (int k0 = 0; k0 < OUT_F; k0 += 4) {  // second 128 K: h_neigh part
        v2f a, b;
        a.x = hnb[mA * OUT_F + k0 + kA];
        a.y = hnb[mA * OUT_F + k0 + kA + 1];
        b.x = W_w[(size_t)n * CAT2_F + IN_F + k0 + kA];
        b.y = W_w[(size_t)n * CAT2_F + IN_F + k0 + kA + 1];
        acc2 = __builtin_amdgcn_wmma_f32_16x16x4_f32(
            false, a, false, b, (short)0, acc2, false, false);
    }

    const float badd = W_b[n] + bias[n];
#pragma unroll
    for (int r = 0; r < 8; ++r) {
        int m = r + ((lane >> 4) << 3);
        out[(size_t)(row0 + m) * OUT_F + n] = acc2[r] + badd;
    }
}

extern "C" void kernel_launch(void* const* d_in, const int* in_sizes, int n_in,
                              void* d_out, int out_size, void* d_ws, size_t ws_size,
                              hipStream_t stream) {
    // setup_inputs order: h, eftr, weight, W_w, W_b, bias, src_idx, dst_idx, num_dst
    const float* h       = (const float*)d_in[0];
    const float* eftr    = (const float*)d_in[1];
    const float* weight  = (const float*)d_in[2];
    const float* W_w     = (const float*)d_in[3];
    const float* W_b     = (const float*)d_in[4];
    const float* bias    = (const float*)d_in[5];
    const int*   src_idx = (const int*)d_in[6];
    const int*   dst_idx = (const int*)d_in[7];
    float*       out     = (float*)d_out;

    // Workspace layout: sum_h | sum_e | cnt  (fp32)
    float* sum_h = (float*)d_ws;
    float* sum_e = sum_h + (size_t)N_NODES * IN_F;
    float* cnt   = sum_e + (size_t)N_NODES * E_F;

    const int ws_elems = N_NODES * (IN_F + E_F + 1);
    zero_ws_kernel<<<(ws_elems + 255) / 256, 256, 0, stream>>>(sum_h, ws_elems);

    // One wave32 per edge, 8 waves per 256-thread block.
    const int scat_blocks = (N_EDGES * 32 + 255) / 256;
    edge_scatter_kernel<<<scat_blocks, 256, 0, stream>>>(
        h, eftr, src_idx, dst_idx, sum_h, sum_e, cnt);

    fused_gemm_kernel<<<N_NODES / 16, 256, 0, stream>>>(
        h, weight, W_w, W_b, bias, sum_h, sum_e, cnt, out);
}